// QuantumLayer_28681791602941
// MI455X (gfx1250) — compile-verified
//
#include <hip/hip_runtime.h>
#include <hip/hip_bf16.h>
#include <math.h>

// ---------------------------------------------------------------------------
// QuantumLayer on MI455X (gfx1250, wave32)
//
// out[b] = <Z0> of  V * psi_enc(x_b)
//   V        : fixed 1024x1024 ansatz unitary (weights only)  -> built once
//   psi_enc  : REAL product state; psi[j] = P0[popc(j&0x2AA)]*P1[popc(j&0x155)]
//
// Kernel 1: register-resident wave32 simulation of the ansatz per basis
//           column -> Vt (row j = column j of V), re/im f32 planes in d_ws.
// Kernel 2: fused  (32768 x 1024) x (1024 x 1024) real GEMM x2 (re, im)
//           using v_wmma_f32_16x16x4_f32, with the |.|^2 sign-reduction
//           folded into the epilogue (Psi never stored).
//
// Needs ws_size >= 8 MB (2 * 1024*1024 * 4 B).
// ---------------------------------------------------------------------------

#define NQ    10
#define NL    6
#define DIM   1024
#define BATCH 32768

typedef __attribute__((ext_vector_type(2))) float v2f;
typedef __attribute__((ext_vector_type(8))) float v8f;

__device__ __forceinline__ float shx(float v, int mask) {
  return __shfl_xor(v, mask, 32);
}

// ---------------------------------------------------------------------------
// Kernel 1: build Vt = ansatz unitary, one wave per basis column.
// Amplitude index a (wire0 = MSB):  lane = a>>5 (wires 0..4), reg = a&31
// (wires 5..9).  wire i<5 <-> lane bit (4-i); wire i>=5 <-> reg bit (9-i).
// ---------------------------------------------------------------------------

template <int I>
__device__ __forceinline__ void apply_rot(float (&re)[32], float (&im)[32],
                                          const int lane, const float* g) {
  const float cx = cosf(0.5f * g[0]), sx = sinf(0.5f * g[0]);
  const float cy = cosf(0.5f * g[1]), sy = sinf(0.5f * g[1]);
  const float cz = cosf(0.5f * g[2]), sz = sinf(0.5f * g[2]);
  if constexpr (I < 5) {
    constexpr int L = 4 - I;
    constexpr int M = 1 << L;
    const int bit = (lane >> L) & 1;
    // RX: new = c*mine - i*s*partner
#pragma unroll
    for (int r = 0; r < 32; ++r) {
      float pre = shx(re[r], M), pim = shx(im[r], M);
      float nr = cx * re[r] + sx * pim;
      float ni = cx * im[r] - sx * pre;
      re[r] = nr; im[r] = ni;
    }
    // RY: bit0: c*mine - s*partner ; bit1: c*mine + s*partner
    const float sys = bit ? sy : -sy;
#pragma unroll
    for (int r = 0; r < 32; ++r) {
      float pre = shx(re[r], M), pim = shx(im[r], M);
      re[r] = cy * re[r] + sys * pre;
      im[r] = cy * im[r] + sys * pim;
    }
    // RZ: phase exp(-+ i t/2) by bit
    const float szs = bit ? sz : -sz;
#pragma unroll
    for (int r = 0; r < 32; ++r) {
      float nr = cz * re[r] - szs * im[r];
      float ni = cz * im[r] + szs * re[r];
      re[r] = nr; im[r] = ni;
    }
  } else {
    constexpr int B = 1 << (9 - I);
#pragma unroll
    for (int r = 0; r < 32; ++r) {
      if ((r & B) == 0) {
        const int r1 = r | B;
        // RX
        float nr0 = cx * re[r]  + sx * im[r1];
        float ni0 = cx * im[r]  - sx * re[r1];
        float nr1 = cx * re[r1] + sx * im[r];
        float ni1 = cx * im[r1] - sx * re[r];
        re[r] = nr0; im[r] = ni0; re[r1] = nr1; im[r1] = ni1;
        // RY
        nr0 = cy * re[r] - sy * re[r1];
        nr1 = sy * re[r] + cy * re[r1];
        ni0 = cy * im[r] - sy * im[r1];
        ni1 = sy * im[r] + cy * im[r1];
        re[r] = nr0; re[r1] = nr1; im[r] = ni0; im[r1] = ni1;
        // RZ
        const float a0 = cz * re[r]  + sz * im[r];
        const float b0 = cz * im[r]  - sz * re[r];
        const float a1 = cz * re[r1] - sz * im[r1];
        const float b1 = cz * im[r1] + sz * re[r1];
        re[r] = a0; im[r] = b0; re[r1] = a1; im[r1] = b1;
      }
    }
  }
}

template <int C>
__device__ __forceinline__ void apply_cnot(float (&re)[32], float (&im)[32],
                                           const int lane) {
  constexpr int T = (C + 1) % NQ;
  if constexpr (C < 4) {                     // lane control, lane target
    constexpr int Lc = 4 - C, Lt = 4 - T;
    const bool ctrl = (lane >> Lc) & 1;
#pragma unroll
    for (int r = 0; r < 32; ++r) {
      float pr = shx(re[r], 1 << Lt);
      float pi = shx(im[r], 1 << Lt);
      re[r] = ctrl ? pr : re[r];
      im[r] = ctrl ? pi : im[r];
    }
  } else if constexpr (C == 4) {             // lane bit0 ctrl, reg bit4 target
    const bool ctrl = lane & 1;
#pragma unroll
    for (int r = 0; r < 16; ++r) {
      const int r1 = r + 16;
      float t0 = re[r], t1 = im[r];
      re[r]  = ctrl ? re[r1] : re[r];
      im[r]  = ctrl ? im[r1] : im[r];
      re[r1] = ctrl ? t0 : re[r1];
      im[r1] = ctrl ? t1 : im[r1];
    }
  } else if constexpr (C < 9) {              // reg ctrl, reg target: free swaps
    constexpr int Rc = 1 << (9 - C), Rt = 1 << (9 - T);
#pragma unroll
    for (int r = 0; r < 32; ++r) {
      if ((r & Rc) && !(r & Rt)) {
        const int r1 = r | Rt;
        float t = re[r]; re[r] = re[r1]; re[r1] = t;
        t = im[r]; im[r] = im[r1]; im[r1] = t;
      }
    }
  } else {                                   // C==9: reg bit0 ctrl, lane bit4 target
#pragma unroll
    for (int r = 1; r < 32; r += 2) {
      re[r] = shx(re[r], 16);
      im[r] = shx(im[r], 16);
    }
  }
}

__global__ __launch_bounds__(256) void build_unitary(
    const float* __restrict__ w, float* __restrict__ vre,
    float* __restrict__ vim) {
  const int lane = threadIdx.x & 31;
  const int col  = (blockIdx.x * blockDim.x + threadIdx.x) >> 5;  // column j
  float re[32], im[32];
  const int clane = col >> 5, creg = col & 31;
#pragma unroll
  for (int r = 0; r < 32; ++r) {
    re[r] = (lane == clane && r == creg) ? 1.0f : 0.0f;
    im[r] = 0.0f;
  }
  for (int l = 0; l < NL; ++l) {
    const float* wl = w + l * NQ * 3;
    apply_rot<0>(re, im, lane, wl + 0);
    apply_rot<1>(re, im, lane, wl + 3);
    apply_rot<2>(re, im, lane, wl + 6);
    apply_rot<3>(re, im, lane, wl + 9);
    apply_rot<4>(re, im, lane, wl + 12);
    apply_rot<5>(re, im, lane, wl + 15);
    apply_rot<6>(re, im, lane, wl + 18);
    apply_rot<7>(re, im, lane, wl + 21);
    apply_rot<8>(re, im, lane, wl + 24);
    apply_rot<9>(re, im, lane, wl + 27);
    apply_cnot<0>(re, im, lane);
    apply_cnot<1>(re, im, lane);
    apply_cnot<2>(re, im, lane);
    apply_cnot<3>(re, im, lane);
    apply_cnot<4>(re, im, lane);
    apply_cnot<5>(re, im, lane);
    apply_cnot<6>(re, im, lane);
    apply_cnot<7>(re, im, lane);
    apply_cnot<8>(re, im, lane);
    apply_cnot<9>(re, im, lane);
  }
  float* orow_r = vre + col * DIM + lane * 32;
  float* orow_i = vim + col * DIM + lane * 32;
#pragma unroll
  for (int r = 0; r < 32; ++r) { orow_r[r] = re[r]; orow_i[r] = im[r]; }
}

// ---------------------------------------------------------------------------
// Kernel 2: fused GEMM + <Z0> reduction.
// Wave owns 16 samples; loops over 16 column-groups (4 N-tiles of 16 each),
// K=1024 in steps of 4 via v_wmma_f32_16x16x4_f32 (re and im planes).
// A (16x4): lane<16: K=0,1 ; lane>=16: K=2,3 ; M = lane&15.
// B (4x16): same K split per half; N = lane&15.
// C (16x16): VGPR v -> M = v + 8*(lane>>4), N = lane&15.
// ---------------------------------------------------------------------------

#define WPB 8   // waves per block
#define SPW 16  // samples per wave

__global__ __launch_bounds__(256) void qgemm_expval(
    const float* __restrict__ x, const float* __restrict__ vre,
    const float* __restrict__ vim, float* __restrict__ out) {
  __shared__ float tab[WPB][SPW][12];
  const int lane = threadIdx.x & 31;
  const int wav  = threadIdx.x >> 5;
  const int b0   = (blockIdx.x * WPB + wav) * SPW;

  if (lane < 16) {  // per-sample encoding tables: P0[k]=c0^(5-k)s0^k, P1 likewise
    const int b = b0 + lane;
    const float h0 = 0.5f * x[2 * b], h1 = 0.5f * x[2 * b + 1];
    const float c0 = cosf(h0), s0 = sinf(h0);
    const float c1 = cosf(h1), s1 = sinf(h1);
    float cp0[6], sp0[6], cp1[6], sp1[6];
    cp0[0] = sp0[0] = cp1[0] = sp1[0] = 1.0f;
#pragma unroll
    for (int k = 1; k < 6; ++k) {
      cp0[k] = cp0[k - 1] * c0; sp0[k] = sp0[k - 1] * s0;
      cp1[k] = cp1[k - 1] * c1; sp1[k] = sp1[k - 1] * s1;
    }
#pragma unroll
    for (int k = 0; k < 6; ++k) {
      tab[wav][lane][k]     = cp0[5 - k] * sp0[k];
      tab[wav][lane][6 + k] = cp1[5 - k] * sp1[k];
    }
  }
  __syncthreads();

  const int half = lane >> 4;
  const int m    = lane & 15;   // M index for A, N index for B/C
  const float* __restrict__ P = &tab[wav][m][0];

  float acc[8];
#pragma unroll
  for (int v = 0; v < 8; ++v) acc[v] = 0.0f;

  for (int g = 0; g < 16; ++g) {  // 4 N-tiles per group: a = g*64 + t*16 + m
    v8f cre[4], cim[4];
#pragma unroll
    for (int t = 0; t < 4; ++t) { cre[t] = (v8f)0.0f; cim[t] = (v8f)0.0f; }

#pragma unroll 2
    for (int k = 0; k < 256; ++k) {
      const int j0 = 4 * k + 2 * half;
      const int j1 = j0 + 1;
      // A operand: psi_b[j] = P0[popc(j&0x2AA)] * P1[popc(j&0x155)]
      v2f A;
      A.x = P[__popc(j0 & 0x2AA)] * P[6 + __popc(j0 & 0x155)];
      A.y = P[__popc(j1 & 0x2AA)] * P[6 + __popc(j1 & 0x155)];
      const float* r0 = vre + j0 * DIM;
      const float* i0 = vim + j0 * DIM;
#pragma unroll
      for (int t = 0; t < 4; ++t) {
        const int a = g * 64 + t * 16 + m;
        v2f Br, Bi;
        Br.x = r0[a]; Br.y = r0[DIM + a];
        Bi.x = i0[a]; Bi.y = i0[DIM + a];
        cre[t] = __builtin_amdgcn_wmma_f32_16x16x4_f32(
            false, A, false, Br, (short)0, cre[t], false, false);
        cim[t] = __builtin_amdgcn_wmma_f32_16x16x4_f32(
            false, A, false, Bi, (short)0, cim[t], false, false);
      }
    }
    const float sgn = (g < 8) ? 1.0f : -1.0f;  // a<512 <=> wire0 bit == 0
#pragma unroll
    for (int t = 0; t < 4; ++t)
#pragma unroll
      for (int v = 0; v < 8; ++v)
        acc[v] += sgn * (cre[t][v] * cre[t][v] + cim[t][v] * cim[t][v]);
  }

  // reduce over N (16 lanes within each half-wave)
#pragma unroll
  for (int v = 0; v < 8; ++v) {
    acc[v] += shx(acc[v], 1);
    acc[v] += shx(acc[v], 2);
    acc[v] += shx(acc[v], 4);
    acc[v] += shx(acc[v], 8);
  }
  if (m == 0) {
#pragma unroll
    for (int v = 0; v < 8; ++v) out[b0 + half * 8 + v] = acc[v];
  }
}

// ---------------------------------------------------------------------------

extern "C" void kernel_launch(void* const* d_in, const int* in_sizes, int n_in,
                              void* d_out, int out_size, void* d_ws,
                              size_t ws_size, hipStream_t stream) {
  const float* x = (const float*)d_in[0];   // (BATCH, 2)
  const float* w = (const float*)d_in[1];   // (NL, NQ, 3)
  float* vre = (float*)d_ws;                // Vt real plane, 4 MB
  float* vim = vre + DIM * DIM;             // Vt imag plane, 4 MB
  (void)in_sizes; (void)n_in; (void)out_size; (void)ws_size;

  build_unitary<<<DIM / WPB, 256, 0, stream>>>(w, vre, vim);
  qgemm_expval<<<BATCH / (WPB * SPW), 256, 0, stream>>>(
      x, vre, vim, (float*)d_out);
}